// GNNCritic4LSTM_8091718386007
// MI455X (gfx1250) — compile-verified
//
#include <hip/hip_runtime.h>
#include <hip/hip_bf16.h>

typedef float v2f __attribute__((ext_vector_type(2)));
typedef float v8f __attribute__((ext_vector_type(8)));

#define NB   8192          // B
#define ACT  6
#define CH   128
#define HID  32
#define NN   (NB*ACT)      // 49152 nodes
#define NE   (NN*16)       // 786432 edges

// ---------------------------------------------------------------------------
// init: zero agg buffer [N*128], set deg[N] = 1.0 (self loop)
// ---------------------------------------------------------------------------
__global__ void init_kernel(float* __restrict__ agg, float* __restrict__ deg) {
    int i = blockIdx.x * blockDim.x + threadIdx.x;
    if (i < NN * CH) agg[i] = 0.f;
    if (i < NN) deg[i] = 1.f;
}

__global__ void deg_count_kernel(const int* __restrict__ dst, float* __restrict__ deg) {
    int e = blockIdx.x * blockDim.x + threadIdx.x;
    if (e < NE) atomicAdd(&deg[dst[e]], 1.f);
}

__global__ void dinv_kernel(float* __restrict__ deg) {
    int i = blockIdx.x * blockDim.x + threadIdx.x;
    if (i < NN) deg[i] = rsqrtf(deg[i]);   // deg >= 1 always
}

// ---------------------------------------------------------------------------
// WMMA f32 GEMM: C[M x 128] = A[M x 128] @ W[:, :128]^T   (W row-major [128 x ldw])
// mode 1 epilogue: += action[row]*W[col][128] + b1[col] + b2[col]
// block = 256 threads (8 waves); block computes 128 rows x 128 cols.
// Weights staged in LDS as K-pairs (float2) so each B fragment is ONE ds_load_b64.
// ---------------------------------------------------------------------------
__global__ __launch_bounds__(256) void gemm_wmma(
    const float* __restrict__ A, const float* __restrict__ W,
    float* __restrict__ C, int ldw, int mode,
    const float* __restrict__ actn, const float* __restrict__ b1,
    const float* __restrict__ b2)
{
    __shared__ float  As[128 * 130];     // row pad 130 -> conflict-free A frags
    __shared__ float2 Wp[64 * 129];      // Wp[k2][n] = {W[n][2k2], W[n][2k2+1]}, pad 129
    const int tid = threadIdx.x;
    const int rowBase = blockIdx.x * 128;

    const float4* Ag = (const float4*)(A + (size_t)rowBase * CH);
    for (int i = tid; i < 128 * 32; i += 256) {
        int r = i >> 5, c4 = i & 31;
        float4 v = Ag[r * 32 + c4];
        float* d = &As[r * 130 + c4 * 4];
        d[0] = v.x; d[1] = v.y; d[2] = v.z; d[3] = v.w;
    }
    for (int i = tid; i < 128 * 64; i += 256) {      // global-coalesced along k2
        int n = i >> 6, k2 = i & 63;
        const float* wr = W + (size_t)n * ldw + 2 * k2;
        Wp[k2 * 129 + n] = make_float2(wr[0], wr[1]);
    }
    __syncthreads();

    const int wave = tid >> 5, lane = tid & 31;
    const int lm = lane & 15, lh = lane >> 4;
    const int m = wave * 16 + lm;

    v8f acc[8] = {};
    for (int kk = 0; kk < 32; ++kk) {
        const int kb = kk * 4 + lh * 2;              // even
        const int k2 = kb >> 1;                      // 2*kk + lh
        v2f a = *(const v2f*)&As[m * 130 + kb];      // one ds_load_b64
#pragma unroll
        for (int t = 0; t < 8; ++t) {
            const int n = t * 16 + lm;
            v2f b = *(const v2f*)&Wp[k2 * 129 + n];  // one ds_load_b64
            acc[t] = __builtin_amdgcn_wmma_f32_16x16x4_f32(
                false, a, false, b, (short)0, acc[t], false, false);
        }
    }
#pragma unroll
    for (int t = 0; t < 8; ++t) {
        const int n = t * 16 + lm;
#pragma unroll
        for (int v = 0; v < 8; ++v) {
            const int r = rowBase + wave * 16 + v + lh * 8;
            float val = acc[t][v];
            if (mode == 1)
                val += actn[r] * W[(size_t)n * ldw + 128] + b1[n] + b2[n];
            C[(size_t)r * CH + n] = val;
        }
    }
}

// ---------------------------------------------------------------------------
// edge message scatter: agg[dst] += xw[src] * dinv[src]*dinv[dst]
// 4 threads per edge, 32 channels each, float4 loads + f32 atomics
// ---------------------------------------------------------------------------
__global__ __launch_bounds__(256) void scatter_kernel(
    const int* __restrict__ src, const int* __restrict__ dst,
    const float* __restrict__ dinv, const float* __restrict__ xw,
    float* __restrict__ agg)
{
    int t = blockIdx.x * blockDim.x + threadIdx.x;
    int e = t >> 2;
    if (e >= NE) return;
    int part = t & 3;
    int s = src[e], d = dst[e];
    float nm = dinv[s] * dinv[d];
    const float4* xp = (const float4*)(xw + (size_t)s * CH) + part * 8;
    float* ap = agg + (size_t)d * CH + part * 32;
#pragma unroll
    for (int i = 0; i < 8; ++i) {
        float4 v = xp[i];
        atomicAdd(ap + i * 4 + 0, v.x * nm);
        atomicAdd(ap + i * 4 + 1, v.y * nm);
        atomicAdd(ap + i * 4 + 2, v.z * nm);
        atomicAdd(ap + i * 4 + 3, v.w * nm);
    }
}

// x = relu(agg + xw/deg + conv_b) + state   (self-loop folded in; in-place on agg)
__global__ void residual_kernel(
    const float* __restrict__ xw, const float* __restrict__ dinv,
    const float* __restrict__ cb, const float* __restrict__ state,
    float* __restrict__ agg)
{
    int i = blockIdx.x * blockDim.x + threadIdx.x;
    if (i >= NN * CH) return;
    int r = i >> 7, c = i & 127;
    float di = dinv[r];
    float v = agg[i] + xw[i] * di * di + cb[c];
    agg[i] = fmaxf(v, 0.f) + state[i];
}

// ---------------------------------------------------------------------------
// Persistent serial LSTM: 1 block, 128 threads (4 waves = 4 gates).
// h,c resident; per step each wave does 2x (8 chained f32 WMMA 16x16x4).
// Weights K-paired in LDS (one b64 per B frag); pre[t+1] register-prefetched.
// ---------------------------------------------------------------------------
__global__ __launch_bounds__(128) void lstm_kernel(
    const float* __restrict__ pre,   // [T*6, 128], gate order i,f,g,o
    const float* __restrict__ w_hh,  // [128, 32]
    float* __restrict__ hs)          // [T*6, 32]
{
    __shared__ float2 Wp[16 * 129];  // Wp[k2][n] = {w_hh[n][2k2], w_hh[n][2k2+1]}
    __shared__ float  hbuf[16 * 34]; // padded batch 6->16, row pad 34 (banks)
    __shared__ float  gact[6 * 128]; // activated gates
    const int tid = threadIdx.x;
    const float2* wh2 = (const float2*)w_hh;         // [128][16] float2, aligned
    for (int i = tid; i < 128 * 16; i += 128) {
        int n = i >> 4, k2 = i & 15;
        Wp[k2 * 129 + n] = wh2[n * 16 + k2];
    }
    for (int i = tid; i < 16 * 34; i += 128) hbuf[i] = 0.f;

    const int wave = tid >> 5, lane = tid & 31;      // wave == gate id
    const int lm = lane & 15, lh = lane >> 4;
    float c0 = 0.f, c1 = 0.f;
    const int e0 = tid, e1 = tid + 128;              // cell-state elements owned
    __syncthreads();

    float pc[2][8];
#pragma unroll
    for (int t2 = 0; t2 < 2; ++t2)
#pragma unroll
        for (int v = 0; v < 8; ++v) {
            int M = v + lh * 8;
            int col = wave * 32 + t2 * 16 + lm;
            pc[t2][v] = (M < ACT) ? pre[(size_t)M * CH + col] : 0.f;
        }

    for (int t = 0; t < NB; ++t) {
        v8f acc0, acc1;
#pragma unroll
        for (int v = 0; v < 8; ++v) { acc0[v] = pc[0][v]; acc1[v] = pc[1][v]; }
        if (t + 1 < NB) {                 // prefetch next step's preactivations
#pragma unroll
            for (int t2 = 0; t2 < 2; ++t2)
#pragma unroll
                for (int v = 0; v < 8; ++v) {
                    int M = v + lh * 8;
                    int col = wave * 32 + t2 * 16 + lm;
                    if (M < ACT) pc[t2][v] = pre[(size_t)((t + 1) * ACT + M) * CH + col];
                }
        }
#pragma unroll
        for (int kk = 0; kk < 8; ++kk) {
            const int kb = kk * 4 + lh * 2;
            const int k2 = kb >> 1;
            v2f a = *(const v2f*)&hbuf[lm * 34 + kb];          // one b64
            const int n0 = wave * 32 + lm;
            v2f b0 = *(const v2f*)&Wp[k2 * 129 + n0];          // one b64
            acc0 = __builtin_amdgcn_wmma_f32_16x16x4_f32(false, a, false, b0, (short)0, acc0, false, false);
            v2f b1 = *(const v2f*)&Wp[k2 * 129 + n0 + 16];     // one b64
            acc1 = __builtin_amdgcn_wmma_f32_16x16x4_f32(false, a, false, b1, (short)0, acc1, false, false);
        }
#pragma unroll
        for (int v = 0; v < 8; ++v) {
            int M = v + lh * 8;
            if (M < ACT) {
                float x0 = acc0[v], x1 = acc1[v];
                if (wave == 2) { x0 = tanhf(x0); x1 = tanhf(x1); }       // g gate
                else { x0 = 1.f / (1.f + expf(-x0)); x1 = 1.f / (1.f + expf(-x1)); }
                gact[M * 128 + wave * 32 + lm] = x0;
                gact[M * 128 + wave * 32 + 16 + lm] = x1;
            }
        }
        __syncthreads();
        {   // c/h update: 192 elements over 128 threads
            int b = e0 >> 5, hh = e0 & 31;
            float gi = gact[b * 128 + hh],      gf = gact[b * 128 + 32 + hh];
            float gg = gact[b * 128 + 64 + hh], go = gact[b * 128 + 96 + hh];
            c0 = gf * c0 + gi * gg;
            float h = go * tanhf(c0);
            hbuf[b * 34 + hh] = h;
            hs[(size_t)(t * ACT + b) * HID + hh] = h;
        }
        if (e1 < ACT * HID) {
            int b = e1 >> 5, hh = e1 & 31;
            float gi = gact[b * 128 + hh],      gf = gact[b * 128 + 32 + hh];
            float gg = gact[b * 128 + 64 + hh], go = gact[b * 128 + 96 + hh];
            c1 = gf * c1 + gi * gg;
            float h = go * tanhf(c1);
            hbuf[b * 34 + hh] = h;
            hs[(size_t)(t * ACT + b) * HID + hh] = h;
        }
        __syncthreads();
    }
}

// ---------------------------------------------------------------------------
// head: out[t] = (sum_b relu(hs[t,b]@lin1^T + lin1_b)) @ lin2^T + lin2_b
// one wave per timestep; lane == hidden unit; shuffle reduction
// ---------------------------------------------------------------------------
__global__ __launch_bounds__(256) void head_kernel(
    const float* __restrict__ hs, const float* __restrict__ l1w,
    const float* __restrict__ l1b, const float* __restrict__ l2w,
    const float* __restrict__ l2b, float* __restrict__ out)
{
    const int wave = threadIdx.x >> 5, h = threadIdx.x & 31;
    const int t = blockIdx.x * 8 + wave;
    float wrow[32];
#pragma unroll
    for (int k = 0; k < 32; ++k) wrow[k] = l1w[h * 32 + k];
    float s = 0.f;
    for (int b = 0; b < ACT; ++b) {
        const float* hp = &hs[(size_t)(t * ACT + b) * HID];
        float d = l1b[h];
#pragma unroll
        for (int k = 0; k < 32; ++k) d += hp[k] * wrow[k];
        s += fmaxf(d, 0.f);
    }
    float v = s * l2w[h];
#pragma unroll
    for (int off = 16; off > 0; off >>= 1) v += __shfl_xor(v, off, 32);
    if (h == 0) out[t] = v + l2b[0];
}

// ---------------------------------------------------------------------------
extern "C" void kernel_launch(void* const* d_in, const int* in_sizes, int n_in,
                              void* d_out, int out_size, void* d_ws, size_t ws_size,
                              hipStream_t stream) {
    const float* state  = (const float*)d_in[0];
    const int*   eidx   = (const int*)d_in[1];
    const float* action = (const float*)d_in[2];
    const float* conv_w = (const float*)d_in[3];
    const float* conv_b = (const float*)d_in[4];
    const float* w_ih   = (const float*)d_in[5];
    const float* w_hh   = (const float*)d_in[6];
    const float* b_ih   = (const float*)d_in[7];
    const float* b_hh   = (const float*)d_in[8];
    const float* lin1_w = (const float*)d_in[9];
    const float* lin1_b = (const float*)d_in[10];
    const float* lin2_w = (const float*)d_in[11];
    const float* lin2_b = (const float*)d_in[12];
    float* out = (float*)d_out;

    const int* src = eidx;
    const int* dst = eidx + NE;

    float* ws   = (float*)d_ws;
    float* deg  = ws;                               // N  (becomes dinv)
    float* bufA = deg + NN;                         // N*128: xw, later pre
    float* bufB = bufA + (size_t)NN * CH;           // N*128: agg, later x
    float* hs   = bufB + (size_t)NN * CH;           // T*6*32

    init_kernel<<<(NN * CH + 255) / 256, 256, 0, stream>>>(bufB, deg);
    deg_count_kernel<<<(NE + 255) / 256, 256, 0, stream>>>(dst, deg);
    dinv_kernel<<<(NN + 255) / 256, 256, 0, stream>>>(deg);

    // xw = state @ conv_w^T
    gemm_wmma<<<NN / 128, 256, 0, stream>>>(state, conv_w, bufA, CH, 0,
                                            nullptr, nullptr, nullptr);
    scatter_kernel<<<(NE * 4 + 255) / 256, 256, 0, stream>>>(src, dst, deg, bufA, bufB);
    residual_kernel<<<(NN * CH + 255) / 256, 256, 0, stream>>>(bufA, deg, conv_b,
                                                               state, bufB);
    // pre = x @ w_ih[:, :128]^T + action (x) w_ih[:,128] + b_ih + b_hh
    gemm_wmma<<<NN / 128, 256, 0, stream>>>(bufB, w_ih, bufA, CH + 1, 1,
                                            action, b_ih, b_hh);
    lstm_kernel<<<1, 128, 0, stream>>>(bufA, w_hh, hs);
    head_kernel<<<NB / 8, 256, 0, stream>>>(hs, lin1_w, lin1_b, lin2_w, lin2_b, out);
}